// RankNetLoss_7060926235074
// MI455X (gfx1250) — compile-verified
//
#include <hip/hip_runtime.h>

// RankNetLoss for B=256, D=256 on gfx1250 (MI455X).
// Compute-bound on softplus; batch reduction done with V_WMMA_F32_16X16X4_F32
// using a ones-row A matrix => exact f32 column sums on the matrix pipe.
// Wave32, EXEC kept all-ones around every WMMA. Transcendentals use the raw
// v_exp_f32 / v_log_f32 hardware ops (arguments are range-safe by
// construction: log arg in [1,2], exp underflow-to-zero is correct).

typedef __attribute__((ext_vector_type(2))) float v2f;
typedef __attribute__((ext_vector_type(8))) float v8f;

#define D_COLS 256
#define B_ROWS 256
#define P_PAIRS (D_COLS * (D_COLS - 1) / 2)   // 32640
#define TILE_PAIRS 16
#define N_TILES (P_PAIRS / TILE_PAIRS)        // 2040
#define WAVES_PER_BLOCK 8
#define BLOCK_THREADS 256
#define N_BLOCKS (N_TILES / WAVES_PER_BLOCK)  // 255

// Invert jnp.triu_indices(D, k=1) enumeration: pairs ordered by i asc, j asc.
// start(i) = i*(2D - i - 1)/2 ; find largest i with start(i) <= p.
__device__ __forceinline__ void pair_from_index(int p, int& oi, int& oj) {
    const float twoDm1 = 2.0f * D_COLS - 1.0f;
    float disc = twoDm1 * twoDm1 - 8.0f * (float)p;
    int i = (int)((twoDm1 - __builtin_sqrtf(disc)) * 0.5f);
    if (i < 0) i = 0;
    if (i > D_COLS - 2) i = D_COLS - 2;
    // fix up float rounding
    while (i > 0 && (i * (2 * D_COLS - i - 1)) / 2 > p) --i;
    while (((i + 1) * (2 * D_COLS - i - 2)) / 2 <= p) ++i;
    oi = i;
    oj = p - (i * (2 * D_COLS - i - 1)) / 2 + i + 1;
}

// softplus(x) = max(x,0) + ln(1 + exp(-|x|))
//             = max(x,0) + ln2 * log2(1 + 2^(-|x| * log2e))
// Raw hardware transcendentals: no denorm/range fixup needed because
//   * 2^(-|x|*log2e) in (0,1]; hardware underflow-to-zero is the right limit
//   * log2 argument in [1,2]: always normal, never huge
__device__ __forceinline__ float softplus_fast(float x) {
    const float LOG2E = 1.4426950408889634f;
    const float LN2   = 0.6931471805599453f;
    float e = __builtin_amdgcn_exp2f(-LOG2E * __builtin_fabsf(x)); // v_exp_f32
    float l = __builtin_amdgcn_logf(1.0f + e);                     // v_log_f32
    return __builtin_fmaxf(x, 0.0f) + LN2 * l;
}

__global__ __launch_bounds__(BLOCK_THREADS)
void ranknet_pair_wmma(const float* __restrict__ pred,
                       const float* __restrict__ rel,
                       float* __restrict__ tile_sum,
                       float* __restrict__ tile_cnt) {
    const int lane = threadIdx.x & 31;
    const int wave = threadIdx.x >> 5;
    const int tile = blockIdx.x * WAVES_PER_BLOCK + wave;   // < N_TILES always
    const int n    = lane & 15;                 // pair-in-tile == B-matrix col N
    const int p    = tile * TILE_PAIRS + n;     // global pair index (< 32640)

    int i, j;
    pair_from_index(p, i, j);

    const float* __restrict__ pi_base = pred + i;
    const float* __restrict__ pj_base = pred + j;
    const float* __restrict__ ri_base = rel + i;
    const float* __restrict__ rj_base = rel + j;

    // A matrix 16x4: row 0 = ones, rows 1..15 = 0  ==> D row 0 = colsum(B).
    // Layout: lane l holds A[l%16][K], v.x: K = (l<16?0:2), v.y: K = (l<16?1:3).
    v2f a;
    a.x = a.y = (n == 0) ? 1.0f : 0.0f;

    v8f accL = {};   // masked loss sums per pair (row M=0, lanes 0..15)
    v8f accC = {};   // valid counts per pair

    // lanes 0-15 supply batch rows {b0, b0+1}; lanes 16-31 supply {b0+2, b0+3}
    const int krow = (lane >> 4) << 1;

#pragma unroll 4
    for (int chunk = 0; chunk < B_ROWS / 4; ++chunk) {
        const int b_lo = chunk * 4 + krow;
        v2f bL, bV;
#pragma unroll
        for (int t = 0; t < 2; ++t) {
            const int off = (b_lo + t) * D_COLS;
            float pi = pi_base[off];
            float pj = pj_base[off];
            float ri = ri_base[off];
            float rj = rj_base[off];
            float s  = (ri > rj) ? 1.0f : ((ri < rj) ? -1.0f : 0.0f);
            float x  = -s * (pi - pj);               // sigma = 1
            float sp = softplus_fast(x);
            float v  = (s != 0.0f) ? 1.0f : 0.0f;
            if (t == 0) { bL.x = sp * v; bV.x = v; }
            else        { bL.y = sp * v; bV.y = v; }
        }
        // D = A*B + C, exact f32 accumulation on the matrix pipe.
        accL = __builtin_amdgcn_wmma_f32_16x16x4_f32(
            false, a, false, bL, (short)0, accL, false, false);
        accC = __builtin_amdgcn_wmma_f32_16x16x4_f32(
            false, a, false, bV, (short)0, accC, false, false);
    }

    // C/D layout: VGPR0, lanes 0..15 -> (M=0, N=lane): our per-pair results.
    float msum = accL[0];
    float cnt  = accC[0];
    float mean    = (cnt > 0.5f) ? (msum / cnt) : 0.0f;
    float contrib = (cnt > 0.5f) ? 1.0f : 0.0f;
    if (lane >= 16) { mean = 0.0f; contrib = 0.0f; }   // M=8 row: defensively zero

    // Deterministic in-wave reduction over the 16 pair lanes.
#pragma unroll
    for (int m = 16; m >= 1; m >>= 1) {
        mean    += __shfl_xor(mean, m, 32);
        contrib += __shfl_xor(contrib, m, 32);
    }
    if (lane == 0) {
        tile_sum[tile] = mean;
        tile_cnt[tile] = contrib;
    }
}

__global__ __launch_bounds__(256)
void ranknet_finalize(const float* __restrict__ tile_sum,
                      const float* __restrict__ tile_cnt,
                      float* __restrict__ out) {
    __shared__ float s1[256];
    __shared__ float s2[256];
    float t = 0.0f, c = 0.0f;
    for (int idx = threadIdx.x; idx < N_TILES; idx += 256) {
        t += tile_sum[idx];
        c += tile_cnt[idx];
    }
    s1[threadIdx.x] = t;
    s2[threadIdx.x] = c;
    __syncthreads();
#pragma unroll
    for (int m = 128; m >= 1; m >>= 1) {
        if (threadIdx.x < m) {
            s1[threadIdx.x] += s1[threadIdx.x + m];
            s2[threadIdx.x] += s2[threadIdx.x + m];
        }
        __syncthreads();
    }
    if (threadIdx.x == 0) {
        float np = s2[0];
        out[0] = (np > 0.5f) ? (s1[0] / np) : 0.0f;
    }
}

extern "C" void kernel_launch(void* const* d_in, const int* in_sizes, int n_in,
                              void* d_out, int out_size, void* d_ws, size_t ws_size,
                              hipStream_t stream) {
    (void)in_sizes; (void)n_in; (void)out_size; (void)ws_size;
    const float* pred = (const float*)d_in[0];   // predictions [256,256] f32
    const float* rel  = (const float*)d_in[1];   // relevance   [256,256] f32

    float* tile_sum = (float*)d_ws;              // N_TILES floats
    float* tile_cnt = tile_sum + 2048;           // aligned second array

    ranknet_pair_wmma<<<N_BLOCKS, BLOCK_THREADS, 0, stream>>>(pred, rel,
                                                              tile_sum, tile_cnt);
    ranknet_finalize<<<1, 256, 0, stream>>>(tile_sum, tile_cnt, (float*)d_out);
}